// MultiGraphEnsembleFC_70806830842517
// MI455X (gfx1250) — compile-verified
//
#include <hip/hip_runtime.h>
#include <hip/hip_bf16.h>

typedef __attribute__((ext_vector_type(2))) float v2f;
typedef __attribute__((ext_vector_type(8))) float v8f;

#if __has_builtin(__builtin_amdgcn_global_load_async_to_lds_b128) && \
    __has_builtin(__builtin_amdgcn_s_wait_asynccnt)
#define USE_ASYNC_LDS 1
#define AS1 __attribute__((address_space(1)))
#define AS3 __attribute__((address_space(3)))
typedef int v4i __attribute__((vector_size(16)));
#endif

__device__ __forceinline__ void atomAddF(float* p, float v) {
    // hardware global_atomic_add_f32 (fp32 atomics are the bandwidth-critical path)
    unsafeAtomicAdd(p, v);
}

// ---------------- utility kernels ----------------
__global__ void fill_f(float* __restrict__ p, float v, int n) {
    int i = blockIdx.x * blockDim.x + threadIdx.x;
    if (i < n) p[i] = v;
}

__global__ void deg_count(const int* __restrict__ dst, float* __restrict__ deg, int E) {
    int i = blockIdx.x * blockDim.x + threadIdx.x;
    if (i < E) atomAddF(&deg[dst[i]], 1.0f);
}

__global__ void rsqrt_ip(float* __restrict__ p, int n) {
    int i = blockIdx.x * blockDim.x + threadIdx.x;
    if (i < n) p[i] = rsqrtf(p[i]);
}

// ---------------- WMMA fp32 GEMM: C[M,Ncol] = A[M,K] @ B[K,Ncol] ----------------
// One wave computes a 16x(16*NT) strip: A panel loaded once per K-step, NT WMMAs
// against NT independent accumulators (A-reuse cuts VMEM traffic ~2x for NT=8).
template <int NT>
__global__ void gemm_wmma_nt(const float* __restrict__ A, const float* __restrict__ B,
                             float* __restrict__ C, int M, int K, int Ncol, int nWaves) {
    int wave = blockIdx.x * (blockDim.x >> 5) + (threadIdx.x >> 5);
    if (wave >= nWaves) return;                // wave-uniform: EXEC all-ones inside
    int lane = threadIdx.x & 31;
    int ngroups = Ncol / (16 * NT);
    int mt = wave / ngroups;
    int ng = wave - mt * ngroups;

    int rowA = mt * 16 + (lane & 15);          // A: lanes 0-15 & 16-31 both hold M=0..15
    int col0 = ng * 16 * NT + (lane & 15);     // B/C: N striped across lane&15
    int kh   = (lane >> 4) << 1;               // lanes>=16 hold K=2,3 within each step

    const float* arow = A + (size_t)rowA * K;
    v8f acc[NT];
#pragma unroll
    for (int j = 0; j < NT; ++j) acc[j] = (v8f){0.f, 0.f, 0.f, 0.f, 0.f, 0.f, 0.f, 0.f};

    for (int k = 0; k < K; k += 4) {
        v2f a;
        a.x = arow[k + kh];
        a.y = arow[k + kh + 1];
        const float* b0 = B + (size_t)(k + kh) * Ncol + col0;
        const float* b1 = B + (size_t)(k + kh + 1) * Ncol + col0;
#pragma unroll
        for (int j = 0; j < NT; ++j) {
            v2f b;
            b.x = b0[j * 16];
            b.y = b1[j * 16];
            acc[j] = __builtin_amdgcn_wmma_f32_16x16x4_f32(false, a, false, b, (short)0,
                                                           acc[j], false, false);
        }
    }
    int m0 = mt * 16 + ((lane >> 4) << 3);     // C: lanes>=16 hold M=8..15
#pragma unroll
    for (int j = 0; j < NT; ++j)
#pragma unroll
        for (int r = 0; r < 8; ++r)
            C[(size_t)(m0 + r) * Ncol + col0 + j * 16] = acc[j][r];
}

// ---------------- GCN edge scatter: acc[d] += h[s] * dinv[s]*dinv[d] ----------------
// thread = (edge, 4-float chunk); float4 gather + 4 hw fp32 atomics
__global__ void scatter_edges(const float* __restrict__ h, const int* __restrict__ src,
                              const int* __restrict__ dst, const float* __restrict__ dinv,
                              float* __restrict__ acc, int E, int H) {
    int idx = blockIdx.x * blockDim.x + threadIdx.x;
    int chunks = H >> 2;
    int e = idx / chunks;
    if (e >= E) return;
    int ch = idx - e * chunks;
    int s = src[e], d = dst[e];
    float nrm = dinv[s] * dinv[d];
    float4 v = *(const float4*)(h + (size_t)s * H + ch * 4);
    float* o = acc + (size_t)d * H + ch * 4;
    atomAddF(o + 0, v.x * nrm);
    atomAddF(o + 1, v.y * nrm);
    atomAddF(o + 2, v.z * nrm);
    atomAddF(o + 3, v.w * nrm);
}

// out[v][c] = maybe_relu(acc[v][c] + h[v][c]*dinv[v]^2 + bias[c])  (self-loop + bias)
__global__ void finalize_prop(const float* __restrict__ acc, const float* __restrict__ h,
                              const float* __restrict__ dinv, const float* __restrict__ bias,
                              float* __restrict__ out, int N, int H, int do_relu) {
    int i = blockIdx.x * blockDim.x + threadIdx.x;
    if (i >= N * H) return;
    int v = i / H;
    int c = i - v * H;
    float dv = dinv[v];
    float r = acc[i] + h[i] * dv * dv + bias[c];
    out[i] = do_relu ? (r > 0.f ? r : 0.f) : r;
}

// ---------------- decoder: 16 edges per wave, 1 wave per block ----------------
// feats[t, e*16+d] = embs[e][i_t][d] + embs[e][j_t][d]; MLP 128->32->16->1, sigmoid
__global__ void decoder_kernel(const float* __restrict__ embs, const int* __restrict__ sl_pos,
                               const int* __restrict__ sl_neg, int EP, int EN, int N,
                               const float* __restrict__ l1w, const float* __restrict__ l1b,
                               const float* __restrict__ l2w, const float* __restrict__ l2b,
                               const float* __restrict__ l3w, const float* __restrict__ l3b,
                               float* __restrict__ out, int Etot) {
    __shared__ __align__(16) float feat[16][128];   // gathered i-endpoint features
#ifdef USE_ASYNC_LDS
    __shared__ __align__(16) float stgB[16][128];   // async staging for j-endpoint
#endif
    __shared__ float h1s[16][32];
    __shared__ float h2s[16][16];
    int lane = threadIdx.x;
    int tile = blockIdx.x;

    // phase 1: build 16x128 feature tile in LDS (embs is L2-resident, random rows)
    for (int idx = lane; idx < 16 * 32; idx += 32) {
        int t   = idx >> 5;
        int r   = idx & 31;
        int enc = r >> 2;
        int dch = r & 3;
        int edge = tile * 16 + t;
        if (edge >= Etot) edge = Etot - 1;     // clamp (Etot is a multiple of 16 anyway)
        int gi, gj;
        if (edge < EP) { gi = sl_pos[edge];       gj = sl_pos[EP + edge]; }
        else           { gi = sl_neg[edge - EP];  gj = sl_neg[EN + edge - EP]; }
        const float* pa = embs + ((size_t)enc * N + gi) * 16 + dch * 4;
        const float* pb = embs + ((size_t)enc * N + gj) * 16 + dch * 4;
#ifdef USE_ASYNC_LDS
        // per-lane global->LDS DMA (16B each), no VGPR round-trip; tracked by ASYNCcnt
        __builtin_amdgcn_global_load_async_to_lds_b128(
            (AS1 v4i*)pa, (AS3 v4i*)&feat[t][enc * 16 + dch * 4], 0, 0);
        __builtin_amdgcn_global_load_async_to_lds_b128(
            (AS1 v4i*)pb, (AS3 v4i*)&stgB[t][enc * 16 + dch * 4], 0, 0);
#else
        float4 va = *(const float4*)pa;
        float4 vb = *(const float4*)pb;
        float4* po = (float4*)&feat[t][enc * 16 + dch * 4];
        *po = make_float4(va.x + vb.x, va.y + vb.y, va.z + vb.z, va.w + vb.w);
#endif
    }
#ifdef USE_ASYNC_LDS
    __builtin_amdgcn_s_wait_asynccnt(0);
    // merge j-endpoint into feat (in-LDS float4 adds)
    for (int idx = lane; idx < 16 * 32; idx += 32) {
        float4* pf = (float4*)&feat[0][0] + idx;
        const float4* pg = (const float4*)&stgB[0][0] + idx;
        float4 a = *pf, b = *pg;
        *pf = make_float4(a.x + b.x, a.y + b.y, a.z + b.z, a.w + b.w);
    }
#endif
    __syncthreads();

    // phase 2: layer1 [16x128]@[128x32] via WMMA, A from LDS
    int kh = (lane >> 4) << 1;
    int rowsel = (lane >> 4) << 3;
    int cl = lane & 15;
    for (int nt = 0; nt < 2; ++nt) {
        v8f acc = {0.f, 0.f, 0.f, 0.f, 0.f, 0.f, 0.f, 0.f};
        int col = nt * 16 + cl;
        for (int k = 0; k < 128; k += 4) {
            v2f a, b;
            a.x = feat[cl][k + kh];
            a.y = feat[cl][k + kh + 1];
            b.x = l1w[(k + kh) * 32 + col];
            b.y = l1w[(k + kh + 1) * 32 + col];
            acc = __builtin_amdgcn_wmma_f32_16x16x4_f32(false, a, false, b, (short)0, acc,
                                                        false, false);
        }
        float bb = l1b[col];
#pragma unroll
        for (int r = 0; r < 8; ++r) {
            float v = acc[r] + bb;
            h1s[rowsel + r][col] = v > 0.f ? v : 0.f;
        }
    }
    __syncthreads();

    // phase 3: layer2 32->16 + relu
    for (int idx = lane; idx < 256; idx += 32) {
        int t = idx >> 4;
        int oo = idx & 15;
        float a = l2b[oo];
#pragma unroll
        for (int k = 0; k < 32; ++k) a += h1s[t][k] * l2w[k * 16 + oo];
        h2s[t][oo] = a > 0.f ? a : 0.f;
    }
    __syncthreads();

    // phase 4: layer3 16->1 + sigmoid
    if (lane < 16) {
        int edge = tile * 16 + lane;
        if (edge < Etot) {
            float a = l3b[0];
#pragma unroll
            for (int k = 0; k < 16; ++k) a += h2s[lane][k] * l3w[k];
            out[edge] = 1.0f / (1.0f + __expf(-a));
        }
    }
}

// ---------------- host orchestration ----------------
extern "C" void kernel_launch(void* const* d_in, const int* in_sizes, int n_in,
                              void* d_out, int out_size, void* d_ws, size_t ws_size,
                              hipStream_t stream) {
    const float* x      = (const float*)d_in[0];
    const int*   sl_pos = (const int*)d_in[1];
    const int*   sl_neg = (const int*)d_in[2];
    const int*   kg     = (const int*)d_in[3];
    const float* W1s    = (const float*)d_in[4];
    const float* b1s    = (const float*)d_in[5];
    const float* W2s    = (const float*)d_in[6];
    const float* b2s    = (const float*)d_in[7];
    const float* l1w    = (const float*)d_in[8];
    const float* l1b    = (const float*)d_in[9];
    const float* l2w    = (const float*)d_in[10];
    const float* l2b    = (const float*)d_in[11];
    const float* l3w    = (const float*)d_in[12];
    const float* l3b    = (const float*)d_in[13];

    const int D_IN = 256, H1 = 128, H2 = 16;
    const int N  = in_sizes[0] / D_IN;
    const int EP = in_sizes[1] / 2;
    const int EN = in_sizes[2] / 2;
    const int EK = in_sizes[3] / 14;      // 7 graphs * 2 rows

    // workspace carving (floats)
    float* ws = (float*)d_ws;
    size_t off = 0;
    float* dinv = ws + off; off += (size_t)N;
    float* xw   = ws + off; off += (size_t)N * H1;
    float* acc1 = ws + off; off += (size_t)N * H1;
    float* h1b  = ws + off; off += (size_t)N * H1;
    float* embs = ws + off; off += (size_t)8 * N * H2;
    float* hw   = acc1;                   // reuse acc1 region after layer-1 finalize
    float* acc2 = acc1 + (size_t)N * H2;

    const int TB = 256;

    for (int e = 0; e < 8; ++e) {
        const int* src;
        const int* dst;
        int E;
        if (e == 0) { src = sl_pos; dst = sl_pos + EP; E = EP; }
        else        { src = kg + (size_t)(e - 1) * 2 * EK; dst = src + EK; E = EK; }
        const float* W1 = W1s + (size_t)e * D_IN * H1;
        const float* b1 = b1s + (size_t)e * H1;
        const float* W2 = W2s + (size_t)e * H1 * H2;
        const float* b2 = b2s + (size_t)e * H2;

        // degree -> dinv
        fill_f<<<(N + TB - 1) / TB, TB, 0, stream>>>(dinv, 1.0f, N);
        deg_count<<<(E + TB - 1) / TB, TB, 0, stream>>>(dst, dinv, E);
        rsqrt_ip<<<(N + TB - 1) / TB, TB, 0, stream>>>(dinv, N);

        // layer 1: XW = x @ W1   [N,256]x[256,128], 8 N-tiles per wave (A reuse)
        int waves1 = N / 16;
        gemm_wmma_nt<8><<<(waves1 + 3) / 4, 128, 0, stream>>>(x, W1, xw, N, D_IN, H1, waves1);
        int n1 = N * H1;
        fill_f<<<(n1 + TB - 1) / TB, TB, 0, stream>>>(acc1, 0.0f, n1);
        int t1 = E * (H1 / 4);
        scatter_edges<<<(t1 + TB - 1) / TB, TB, 0, stream>>>(xw, src, dst, dinv, acc1, E, H1);
        finalize_prop<<<(n1 + TB - 1) / TB, TB, 0, stream>>>(acc1, xw, dinv, b1, h1b, N, H1, 1);

        // layer 2: HW = H1 @ W2  [N,128]x[128,16]
        int waves2 = N / 16;
        gemm_wmma_nt<1><<<(waves2 + 3) / 4, 128, 0, stream>>>(h1b, W2, hw, N, H1, H2, waves2);
        int n2 = N * H2;
        fill_f<<<(n2 + TB - 1) / TB, TB, 0, stream>>>(acc2, 0.0f, n2);
        int t2 = E * (H2 / 4);
        scatter_edges<<<(t2 + TB - 1) / TB, TB, 0, stream>>>(hw, src, dst, dinv, acc2, E, H2);
        finalize_prop<<<(n2 + TB - 1) / TB, TB, 0, stream>>>(acc2, hw, dinv, b2,
                                                             embs + (size_t)e * N * H2, N, H2, 0);
    }

    // decoder over concat(sl_pos, sl_neg)
    int Etot = EP + EN;
    int tiles = (Etot + 15) / 16;
    decoder_kernel<<<tiles, 32, 0, stream>>>(embs, sl_pos, sl_neg, EP, EN, N,
                                             l1w, l1b, l2w, l2b, l3w, l3b,
                                             (float*)d_out, Etot);
}